// Exp_41506563948871
// MI455X (gfx1250) — compile-verified
//
#include <hip/hip_runtime.h>
#include <stdint.h>

// Field: (B=2, C=3, X=128, Y=128, Z=128), float32, z contiguous.
#define DIM    128
#define NCH    3
#define NB     2
#define VOXSH  21              // 128^3 = 2^21 elements per (b,c) plane

// Output tile per block: 8 x 8 x 16 (1024 voxels, 4 per thread).
// Halo: +-2 in x/y, +-4 in z (z chunk starts stay multiples of 4 since
// tz0 % 16 == 0, so every B128 chunk is 16B-aligned and never straddles
// the wrap seam).
#define TX 8
#define TY 8
#define TZ 16
#define RXY 2
#define RZ  4
#define SX  (TX + 2*RXY)             // 12
#define SY  (TY + 2*RXY)             // 12
#define SZ  (TZ + 2*RZ)              // 24
#define CH_STRIDE   (SX*SY*SZ)       // 3456 floats per channel
#define STAGE_FLOATS (NCH*CH_STRIDE) // 10368 floats = 41472 B LDS
#define ZCHUNKS     (SZ/4)           // 6 B128 chunks per z-row
#define STAGE_CHUNKS (STAGE_FLOATS/4)// 2592 B128 chunks

// phi0 = v * 2^-8, elementwise, float4-vectorized (counts divisible by 4).
__global__ __launch_bounds__(256) void init_scale(const float* __restrict__ v,
                                                  float* __restrict__ phi, int n4) {
    int i = blockIdx.x * 256 + threadIdx.x;
    if (i < n4) {
        float4 t = reinterpret_cast<const float4*>(v)[i];
        t.x *= 0.00390625f; t.y *= 0.00390625f;
        t.z *= 0.00390625f; t.w *= 0.00390625f;
        reinterpret_cast<float4*>(phi)[i] = t;
    }
}

// One squaring step: dst(x) = src(x) + trilerp_wrap(src, x + src(x)).
__global__ __launch_bounds__(256) void square_step(const float* __restrict__ src,
                                                   float* __restrict__ dst) {
    __shared__ __align__(16) float sA[STAGE_FLOATS];

    const int bid = blockIdx.x;
    const int b   = bid >> 11;                  // 2048 tiles per batch
    const int t   = bid & 2047;
    const int tz0 = (t & 7) << 4;               // 8 z-tiles
    const int ty0 = ((t >> 3) & 15) << 3;       // 16 y-tiles
    const int tx0 = (t >> 7) << 3;              // 16 x-tiles
    const int tid = threadIdx.x;
    const int bch = b * NCH;

    // ---- Kick off async staging of the (12,12,24) x 3ch halo window.
    const uint64_t srcBase = (uint64_t)(uintptr_t)src;
    for (int ci = tid; ci < STAGE_CHUNKS; ci += 256) {
        int zc  = ci % ZCHUNKS;
        int rem = ci / ZCHUNKS;
        int ly  = rem % SY; rem /= SY;
        int lx  = rem % SX;
        int c   = rem / SX;
        int gx  = (tx0 - RXY + lx)  & (DIM - 1);
        int gy  = (ty0 - RXY + ly)  & (DIM - 1);
        int gz  = (tz0 - RZ + 4*zc) & (DIM - 1);    // chunk start, mult of 4
        uint64_t gaddr = srcBase +
            4ull * ((((uint64_t)(bch + c) << 7 | (unsigned)gx) << 7 | (unsigned)gy) << 7
                    | (unsigned)gz);
        unsigned ldsAddr =
            (unsigned)(uintptr_t)(void*)&sA[((c * SX + lx) * SY + ly) * SZ + 4*zc];
        asm volatile("global_load_async_to_lds_b128 %0, %1, off"
                     :: "v"(ldsAddr), "v"(gaddr) : "memory");
    }

    // ---- Overlap: prefetch this thread's own phi(x) via normal global
    // loads (LOADcnt, L2-resident) while the async engine fills LDS.
    const int oz  = tid & 15;
    const int oy  = (tid >> 4) & 7;
    const int oxq = tid >> 7;                   // 0..1
    const int y = ty0 + oy, z = tz0 + oz;
    const uint64_t ob = (uint64_t)bch << VOXSH;

    float pxv[4], pyv[4], pzv[4];
#pragma unroll
    for (int h = 0; h < 4; ++h) {
        const int x = tx0 + oxq * 4 + h;
        const uint64_t vo = ob + ((uint64_t)x << 14) + ((unsigned)y << 7) + (unsigned)z;
        pxv[h] = src[vo];
        pyv[h] = src[vo + (1u << VOXSH)];
        pzv[h] = src[vo + (2u << VOXSH)];
    }

    asm volatile("s_wait_asynccnt 0x0" ::: "memory");
    __syncthreads();

    // ---- Compute 4 voxels per thread; z-fastest lanes give 64B-contiguous
    // stores per wave per channel.
#pragma unroll
    for (int h = 0; h < 4; ++h) {
        const int ox = oxq * 4 + h;
        const int x  = tx0 + ox;
        const float px = pxv[h], py = pyv[h], pz = pzv[h];

        const float fx = (float)x + px, fy = (float)y + py, fz = (float)z + pz;
        const float flx = floorf(fx), fly = floorf(fy), flz = floorf(fz);
        const float wx = fx - flx,  wy = fy - fly,  wz = fz - flz;
        const int ix = (int)flx, iy = (int)fly, iz = (int)flz;

        const float wxs[2] = {1.0f - wx, wx};
        const float wys[2] = {1.0f - wy, wy};
        const float wzs[2] = {1.0f - wz, wz};

        float ax = 0.0f, ay = 0.0f, az = 0.0f;

        const int lx0 = ix - (tx0 - RXY);
        const int ly0 = iy - (ty0 - RXY);
        const int lz0 = iz - (tz0 - RZ);
        const bool inwin = ((unsigned)lx0 <= (unsigned)(SX - 2)) &&
                           ((unsigned)ly0 <= (unsigned)(SY - 2)) &&
                           ((unsigned)lz0 <= (unsigned)(SZ - 2));

        if (inwin) {                            // fast path: LDS gathers
            const int base = (lx0 * SY + ly0) * SZ + lz0;
#pragma unroll
            for (int dx = 0; dx < 2; ++dx)
#pragma unroll
                for (int dy = 0; dy < 2; ++dy)
#pragma unroll
                    for (int dz = 0; dz < 2; ++dz) {   // dz pairs -> ds_load_b64
                        const float w = wxs[dx] * wys[dy] * wzs[dz];
                        const int o  = base + dx * (SY * SZ) + dy * SZ + dz;
                        ax += w * sA[o];
                        ay += w * sA[o + CH_STRIDE];
                        az += w * sA[o + 2 * CH_STRIDE];
                    }
        } else {                                // rare tail: global gather w/ wrap
            const int gxs[2] = { ix & (DIM-1), (ix + 1) & (DIM-1) };
            const int gys[2] = { iy & (DIM-1), (iy + 1) & (DIM-1) };
            const int gzs[2] = { iz & (DIM-1), (iz + 1) & (DIM-1) };
#pragma unroll
            for (int dx = 0; dx < 2; ++dx)
#pragma unroll
                for (int dy = 0; dy < 2; ++dy)
#pragma unroll
                    for (int dz = 0; dz < 2; ++dz) {
                        const float w = wxs[dx] * wys[dy] * wzs[dz];
                        const uint64_t o = ob + ((uint64_t)gxs[dx] << 14)
                                              + ((unsigned)gys[dy] << 7)
                                              + (unsigned)gzs[dz];
                        ax += w * src[o];
                        ay += w * src[o + (1u << VOXSH)];
                        az += w * src[o + (2u << VOXSH)];
                    }
        }

        const uint64_t vo = ob + ((uint64_t)x << 14) + ((unsigned)y << 7) + (unsigned)z;
        dst[vo]                   = px + ax;
        dst[vo + (1u << VOXSH)]   = py + ay;
        dst[vo + (2u << VOXSH)]   = pz + az;
    }
}

extern "C" void kernel_launch(void* const* d_in, const int* in_sizes, int n_in,
                              void* d_out, int out_size, void* d_ws, size_t ws_size,
                              hipStream_t stream) {
    const float* v = (const float*)d_in[0];
    float* out = (float*)d_out;
    float* ws  = (float*)d_ws;                  // needs >= 50.3 MB scratch

    const int n  = NB * NCH * DIM * DIM * DIM;  // 12,582,912 == out_size
    const int n4 = n / 4;
    init_scale<<<(n4 + 255) / 256, 256, 0, stream>>>(v, out, n4);

    float* bufs[2] = { out, ws };
    int cur = 0;
    for (int s = 0; s < 8; ++s) {               // even #steps -> ends in d_out
        square_step<<<NB * 2048, 256, 0, stream>>>(bufs[cur], bufs[cur ^ 1]);
        cur ^= 1;
    }
    (void)in_sizes; (void)n_in; (void)out_size; (void)ws_size;
}